// GNNPolicy_43654047597026
// MI455X (gfx1250) — compile-verified
//
#include <hip/hip_runtime.h>
#include <hip/hip_bf16.h>

// ---------------------------------------------------------------------------
// GNNPolicy for MI455X (gfx1250, wave32, WMMA).
//
// Math notes (exact simplifications, not approximations):
//  * LayerNorm over a 1-wide feature axis returns its bias exactly
//    => constraint embedding c is ONE constant 32-vector, edge embedding is a
//       constant scalar. cf / ea inputs are mathematically dead.
//  * conv(cons<-var): relu(segsum(msg_v) + deg_c*et + c0@Wself+b)
//  * conv(var<-cons): relu(deg_v*(c0@Wmsg+b + et) + v@Wself+b)
//
// Heavy parts:
//  * per-variable MLP 6->32->32->(17 msg + 17 self): done with
//    V_WMMA_F32_16X16X4_F32, one 16-node tile per wave, GEMM stages chained
//    through per-wave LDS (keeps f32 precision; problem is memory-bound so
//    low-precision WMMA would gain nothing).
//  * 3M-edge gather/scatter: msg table + accumulators are ~24MB each -> L2
//    resident on the 192MB L2; per-edge f32 global atomics.
// ---------------------------------------------------------------------------

typedef float v2f __attribute__((ext_vector_type(2)));
typedef float v8f __attribute__((ext_vector_type(8)));

#define MSTR 20          // padded row stride (floats) for 17-ch rows (80B, 16B aligned)
#define MAXB 16          // max graphs supported in pooling buffers
#define LN_EPS 1e-5f

// consts region layout (float offsets)
#define C_C0    0
#define C_ET    32
#define C_CMSG  64
#define C_CSELF 96
#define C_BC48  128
#define C_W1P   256      // [8][32]  padded var_lin1.W^ (K-major)
#define C_WCP   512      // [32][48] combined msg+self weights (K-major)
#define C_TOTAL 2048

// -------------------------- prep: fold constants ---------------------------
__global__ __launch_bounds__(256) void kprep(
    const float* __restrict__ cons_ln_b,
    const float* __restrict__ c1W, const float* __restrict__ c1b,
    const float* __restrict__ c2W, const float* __restrict__ c2b,
    const float* __restrict__ edge_ln_b,
    const float* __restrict__ m0W, const float* __restrict__ m0b,
    const float* __restrict__ s0W, const float* __restrict__ s0b,
    const float* __restrict__ e0W, const float* __restrict__ e0b,
    const float* __restrict__ m1W, const float* __restrict__ m1b,
    const float* __restrict__ s1W, const float* __restrict__ s1b,
    const float* __restrict__ e1W, const float* __restrict__ e1b,
    const float* __restrict__ m2W, const float* __restrict__ m2b,
    const float* __restrict__ s2W, const float* __restrict__ s2b,
    const float* __restrict__ e2W, const float* __restrict__ e2b,
    const float* __restrict__ vl1W,
    float* __restrict__ consts)
{
  __shared__ float a1[32];
  __shared__ float c0[32];
  int t = threadIdx.x;
  float clnb = cons_ln_b[0];
  float elnb = edge_ln_b[0];
  if (t < 32) a1[t] = fmaxf(clnb * c1W[t] + c1b[t], 0.f);
  __syncthreads();
  if (t < 32) {
    float s = c2b[t];
    for (int k = 0; k < 32; ++k) s += a1[k] * c2W[k * 32 + t];
    c0[t] = fmaxf(s, 0.f);
    consts[C_C0 + t] = c0[t];
  }
  __syncthreads();
  if (t < 32) {
    if (t < 17) {
      const float *mW, *mb, *sW, *sb, *eW, *eb; int h, hd;
      if (t < 8)       { mW=m0W; mb=m0b; sW=s0W; sb=s0b; eW=e0W; eb=e0b; h=t;    hd=8; }
      else if (t < 16) { mW=m1W; mb=m1b; sW=s1W; sb=s1b; eW=e1W; eb=e1b; h=t-8;  hd=8; }
      else             { mW=m2W; mb=m2b; sW=s2W; sb=s2b; eW=e2W; eb=e2b; h=0;    hd=1; }
      consts[C_ET + t] = elnb * eW[h] + eb[h];
      float sm = mb[h], ss = sb[h];
      for (int k = 0; k < 32; ++k) {
        sm += c0[k] * mW[k * hd + h];
        ss += c0[k] * sW[k * hd + h];
      }
      consts[C_CMSG + t]  = sm;
      consts[C_CSELF + t] = ss;
    } else {
      consts[C_ET + t] = 0.f; consts[C_CMSG + t] = 0.f; consts[C_CSELF + t] = 0.f;
    }
  }
  if (t < 48) {
    float v = 0.f;
    if (t < 8)       v = m0b[t];
    else if (t < 16) v = m1b[t - 8];
    else if (t == 16) v = m2b[0];
    else if (t < 25) v = s0b[t - 17];
    else if (t < 33) v = s1b[t - 25];
    else if (t == 33) v = s2b[0];
    consts[C_BC48 + t] = v;
  }
  // W1p: [8][32], rows 6..7 zero
  for (int idx = t; idx < 256; idx += 256) {
    int k = idx >> 5, j = idx & 31;
    consts[C_W1P + idx] = (k < 6) ? vl1W[k * 32 + j] : 0.f;
  }
  // Wcp: [32][48] combined (msg0|msg1|msg2|self0|self1|self2|pad)
  for (int idx = t; idx < 1536; idx += 256) {
    int k = idx / 48, m = idx % 48;
    float v = 0.f;
    if (m < 8)        v = m0W[k * 8 + m];
    else if (m < 16)  v = m1W[k * 8 + (m - 8)];
    else if (m == 16) v = m2W[k];
    else if (m < 25)  v = s0W[k * 8 + (m - 17)];
    else if (m < 33)  v = s1W[k * 8 + (m - 25)];
    else if (m == 33) v = s2W[k];
    consts[C_WCP + idx] = v;
  }
}

// ------------------- pass A: global max of |obj coeff| ---------------------
__global__ __launch_bounds__(256) void kgmax(const float* __restrict__ vf,
                                             unsigned* __restrict__ gmax, int nv)
{
  __shared__ float red[256];
  int i = blockIdx.x * 256 + threadIdx.x;
  float val = 0.f;
  if (i < nv) {
    float f0 = vf[(size_t)i * 6 + 0];
    float f1 = vf[(size_t)i * 6 + 1];
    float f2 = vf[(size_t)i * 6 + 2];
    float cfac = fmaxf(fmaxf(fabsf(f0), fabsf(f1)), 1.0f);
    val = fabsf(f2) * cfac;
  }
  red[threadIdx.x] = val;
  __syncthreads();
  for (int s = 128; s > 0; s >>= 1) {
    if (threadIdx.x < s) red[threadIdx.x] = fmaxf(red[threadIdx.x], red[threadIdx.x + s]);
    __syncthreads();
  }
  if (threadIdx.x == 0) atomicMax(gmax, __float_as_uint(red[0]));  // vals >= 0
}

// ---------------- WMMA GEMM stage helper (one 16-node tile) ----------------
// Computes D^T = W^T(tile) x Xt : A = W^T tile (f32 16x4 operand layout),
// B = Xt[K][16] from LDS.  W is row-major [Ktot][Ms].
template <int MT, int KT>
__device__ __forceinline__ void gemm_stage(const float* __restrict__ W, int Ms,
                                           const float* __restrict__ bias,
                                           const float* __restrict__ xt,
                                           int nloc, int khalf, int mhalf,
                                           v8f out[MT])
{
#pragma unroll
  for (int mt = 0; mt < MT; ++mt) {
    v8f c;
#pragma unroll
    for (int r = 0; r < 8; ++r) c[r] = bias[mt * 16 + mhalf + r];
#pragma unroll
    for (int kk = 0; kk < KT; ++kk) {
      int kb = kk * 4 + khalf;
      v2f a, b;
      a[0] = W[(kb + 0) * Ms + mt * 16 + nloc];
      a[1] = W[(kb + 1) * Ms + mt * 16 + nloc];
      b[0] = xt[(kb + 0) * 16 + nloc];
      b[1] = xt[(kb + 1) * 16 + nloc];
      c = __builtin_amdgcn_wmma_f32_16x16x4_f32(false, a, false, b, (short)0, c,
                                                false, false);
    }
    out[mt] = c;
  }
}

#define LDS_FENCE() do { asm volatile("s_wait_dscnt 0x0" ::: "memory"); \
                         __builtin_amdgcn_wave_barrier(); } while (0)

// ---------------- pass B: variable MLP -> msg[17], self[17] ----------------
__global__ __launch_bounds__(256) void kvarmlp(
    const float* __restrict__ vf,
    const float* __restrict__ lng, const float* __restrict__ lnb,
    const float* __restrict__ gmaxp,
    const float* __restrict__ consts,
    const float* __restrict__ b1,
    const float* __restrict__ W2, const float* __restrict__ b2,
    float* __restrict__ msg, float* __restrict__ selfb,
    int nv, int numTiles)
{
  __shared__ float Xt[8][32 * 16];   // per-wave K-major tile buffer (16KB/block)
  int wid  = threadIdx.x >> 5;
  int lane = threadIdx.x & 31;
  int tile = blockIdx.x * 8 + wid;
  if (tile >= numTiles) return;      // wave-uniform
  float* xt = Xt[wid];
  int nloc  = lane & 15;
  int khalf = (lane < 16) ? 0 : 2;   // K offset of the upper lane half (A/B ops)
  int mhalf = (lane < 16) ? 0 : 8;   // M offset of the upper lane half (C/D op)
  float gmax = *gmaxp;

  // phase 1: normalize + LayerNorm(6) -> Xt rows 0..5 (rows 6,7 zero pad)
  if (lane < 16) {
    int i = tile * 16 + nloc;
    float x[6] = {0.f, 0.f, 0.f, 0.f, 0.f, 0.f};
    if (i < nv) {
      float f0 = vf[(size_t)i * 6 + 0], f1 = vf[(size_t)i * 6 + 1];
      float f2 = vf[(size_t)i * 6 + 2], f3 = vf[(size_t)i * 6 + 3];
      float f4 = vf[(size_t)i * 6 + 4], f5 = vf[(size_t)i * 6 + 5];
      float cfac = fmaxf(fmaxf(fabsf(f0), fabsf(f1)), 1.0f);
      x[0] = f0 / cfac; x[1] = f1 / cfac; x[2] = f2 * cfac / gmax;
      x[3] = f3; x[4] = f4; x[5] = f5;
    }
    float mu = (x[0] + x[1] + x[2] + x[3] + x[4] + x[5]) * (1.f / 6.f);
    float var = 0.f;
#pragma unroll
    for (int k = 0; k < 6; ++k) { float d = x[k] - mu; var += d * d; }
    var *= (1.f / 6.f);
    float inv = rsqrtf(var + LN_EPS);
#pragma unroll
    for (int k = 0; k < 6; ++k) xt[k * 16 + nloc] = (x[k] - mu) * inv * lng[k] + lnb[k];
    xt[6 * 16 + nloc] = 0.f;
    xt[7 * 16 + nloc] = 0.f;
  }
  LDS_FENCE();

  // stage 1: h1^T(32x16) = W1p^T x X  (K=8 padded)
  v8f h[2];
  gemm_stage<2, 2>(consts + C_W1P, 32, b1, xt, nloc, khalf, mhalf, h);
#pragma unroll
  for (int mt = 0; mt < 2; ++mt)
#pragma unroll
    for (int r = 0; r < 8; ++r)
      xt[(mt * 16 + mhalf + r) * 16 + nloc] = fmaxf(h[mt][r], 0.f);
  LDS_FENCE();

  // stage 2: v^T(32x16) = W2^T x h1  (K=32)
  v8f vv[2];
  gemm_stage<2, 8>(W2, 32, b2, xt, nloc, khalf, mhalf, vv);
#pragma unroll
  for (int mt = 0; mt < 2; ++mt)
#pragma unroll
    for (int r = 0; r < 8; ++r)
      xt[(mt * 16 + mhalf + r) * 16 + nloc] = fmaxf(vv[mt][r], 0.f);
  LDS_FENCE();

  // stage 3: out^T(48x16) = Wcp^T x v  (ch 0..16 = msg, 17..33 = self, rest pad)
  v8f o[3];
  gemm_stage<3, 8>(consts + C_WCP, 48, consts + C_BC48, xt, nloc, khalf, mhalf, o);
  int node = tile * 16 + nloc;
  if (node < nv) {
#pragma unroll
    for (int mt = 0; mt < 3; ++mt)
#pragma unroll
      for (int r = 0; r < 8; ++r) {
        int ch = mt * 16 + mhalf + r;
        if (ch < 17)       msg[(size_t)node * MSTR + ch]          = o[mt][r];
        else if (ch < 34)  selfb[(size_t)node * MSTR + (ch - 17)] = o[mt][r];
      }
  }
}

// ---------------- pass C: edge scatter (deg + msg accumulation) ------------
__global__ __launch_bounds__(256) void kedges(const int* __restrict__ ei,
                                              const float* __restrict__ msg,
                                              float* __restrict__ aggc,
                                              int* __restrict__ degv,
                                              int* __restrict__ degc, int ne)
{
  int e = blockIdx.x * 256 + threadIdx.x;
  if (e >= ne) return;
  int s = ei[e];
  int d = ei[(size_t)ne + e];
  atomicAdd(&degv[s], 1);
  atomicAdd(&degc[d], 1);
  const float* mr = msg + (size_t)s * MSTR;
  float* ag = aggc + (size_t)d * MSTR;
#pragma unroll
  for (int k = 0; k < 17; ++k) atomicAdd(ag + k, mr[k]);
}

// ---------------- pass D: constraint relu + mean-pool ----------------------
__global__ __launch_bounds__(256) void kcons(const float* __restrict__ aggc,
                                             const int* __restrict__ degc,
                                             const int* __restrict__ cbatch,
                                             const float* __restrict__ consts,
                                             float* __restrict__ poolC,
                                             int* __restrict__ cntC, int nc)
{
  __shared__ float sp[MAXB * 17];
  __shared__ int   sc[MAXB];
  int t = threadIdx.x;
  for (int k = t; k < MAXB * 17; k += 256) sp[k] = 0.f;
  if (t < MAXB) sc[t] = 0;
  __syncthreads();
  int j = blockIdx.x * 256 + t;
  if (j < nc) {
    int b = cbatch[j] & (MAXB - 1);
    float dg = (float)degc[j];
    const float* et = consts + C_ET;
    const float* cs = consts + C_CSELF;
    const float* ar = aggc + (size_t)j * MSTR;
    atomicAdd(&sc[b], 1);
#pragma unroll
    for (int k = 0; k < 17; ++k) {
      float v = fmaxf(ar[k] + dg * et[k] + cs[k], 0.f);
      atomicAdd(&sp[b * 17 + k], v);
    }
  }
  __syncthreads();
  for (int k = t; k < MAXB * 17; k += 256) if (sp[k] != 0.f) atomicAdd(&poolC[k], sp[k]);
  if (t < MAXB && sc[t]) atomicAdd(&cntC[t], sc[t]);
}

// ---------------- pass E: variable relu + mean-pool ------------------------
__global__ __launch_bounds__(256) void kvars(const float* __restrict__ selfb,
                                             const int* __restrict__ degv,
                                             const int* __restrict__ vbatch,
                                             const float* __restrict__ consts,
                                             float* __restrict__ poolV,
                                             int* __restrict__ cntV, int nv)
{
  __shared__ float sp[MAXB * 17];
  __shared__ int   sc[MAXB];
  int t = threadIdx.x;
  for (int k = t; k < MAXB * 17; k += 256) sp[k] = 0.f;
  if (t < MAXB) sc[t] = 0;
  __syncthreads();
  int i = blockIdx.x * 256 + t;
  if (i < nv) {
    int b = vbatch[i] & (MAXB - 1);
    float dg = (float)degv[i];
    const float* et = consts + C_ET;
    const float* cm = consts + C_CMSG;
    const float* sr = selfb + (size_t)i * MSTR;
    atomicAdd(&sc[b], 1);
#pragma unroll
    for (int k = 0; k < 17; ++k) {
      float v = fmaxf(dg * (cm[k] + et[k]) + sr[k], 0.f);
      atomicAdd(&sp[b * 17 + k], v);
    }
  }
  __syncthreads();
  for (int k = t; k < MAXB * 17; k += 256) if (sp[k] != 0.f) atomicAdd(&poolV[k], sp[k]);
  if (t < MAXB && sc[t]) atomicAdd(&cntV[t], sc[t]);
}

// ---------------- final: diff -> LN(34) -> 128 relu -> sigmoid -------------
__global__ __launch_bounds__(128) void kfinal(
    const float* __restrict__ poolVs, const float* __restrict__ poolCs,
    const int* __restrict__ cntVs, const int* __restrict__ cntCs,
    const float* __restrict__ poolVt, const float* __restrict__ poolCt,
    const int* __restrict__ cntVt, const int* __restrict__ cntCt,
    const float* __restrict__ g, const float* __restrict__ bb,
    const float* __restrict__ W1, const float* __restrict__ b1,
    const float* __restrict__ W2, const int* __restrict__ ngp,
    float* __restrict__ out)
{
  __shared__ float xs[34];
  __shared__ float red[128];
  int ng = *ngp;
  int t = threadIdx.x;
  for (int b = 0; b < ng; ++b) {
    if (t < 34) {
      float s0, s1;
      if (t < 17) {
        s0 = poolVs[b * 17 + t] / (float)cntVs[b];
        s1 = poolVt[b * 17 + t] / (float)cntVt[b];
      } else {
        int kk = t - 17;
        s0 = poolCs[b * 17 + kk] / (float)cntCs[b];
        s1 = poolCt[b * 17 + kk] / (float)cntCt[b];
      }
      xs[t] = s1 - s0;      // -score0 + score1
    }
    __syncthreads();
    if (t == 0) {
      float mu = 0.f;
      for (int k = 0; k < 34; ++k) mu += xs[k];
      mu *= (1.f / 34.f);
      float var = 0.f;
      for (int k = 0; k < 34; ++k) { float d = xs[k] - mu; var += d * d; }
      var *= (1.f / 34.f);
      float inv = rsqrtf(var + LN_EPS);
      for (int k = 0; k < 34; ++k) xs[k] = (xs[k] - mu) * inv * g[k] + bb[k];
    }
    __syncthreads();
    float part = 0.f;
    {
      float h = b1[t];
      for (int k = 0; k < 34; ++k) h += xs[k] * W1[k * 128 + t];
      part = fmaxf(h, 0.f) * W2[t];
    }
    red[t] = part;
    __syncthreads();
    for (int s = 64; s > 0; s >>= 1) {
      if (t < s) red[t] += red[t + s];
      __syncthreads();
    }
    if (t == 0) out[b] = 1.f / (1.f + expf(-red[0]));
    __syncthreads();
  }
}

// ---------------------------------------------------------------------------
extern "C" void kernel_launch(void* const* d_in, const int* in_sizes, int n_in,
                              void* d_out, int out_size, void* d_ws, size_t ws_size,
                              hipStream_t stream)
{
  (void)n_in; (void)out_size; (void)ws_size;
  const int NC = in_sizes[0];
  const int NE = in_sizes[2];
  const int NV = in_sizes[3] / 6;

  // ---- input pointers (setup_inputs dict order; params flattened depth-first
  //      in dict-insertion order) ----
  const int*   ngp = (const int*)d_in[12];
  const float* P[64];
  for (int i = 13; i < 50 && i < n_in; ++i) P[i] = (const float*)d_in[i];
  const float* cons_ln_b = P[14];
  const float* c1W = P[15];  const float* c1b = P[16];
  const float* c2W = P[17];  const float* c2b = P[18];
  const float* edge_ln_b = P[20];
  const float* var_ln_g = P[21]; const float* var_ln_b = P[22];
  const float* vl1W = P[23]; const float* vl1b = P[24];
  const float* vl2W = P[25]; const float* vl2b = P[26];
  const float* cv[3][6];
  for (int l = 0; l < 3; ++l)
    for (int a = 0; a < 6; ++a) cv[l][a] = P[27 + 6 * l + a];
  const float* fln_g = P[45]; const float* fln_b = P[46];
  const float* f1W = P[47];   const float* f1b = P[48];
  const float* f2W = P[49];

  // ---- workspace carve-up (256B aligned chunks) ----
  size_t off = 0;
  auto give = [&](size_t bytes) -> char* {
    char* p = (char*)d_ws + off;
    off += (bytes + 255) & ~(size_t)255;
    return p;
  };
  unsigned* gmaxp  = (unsigned*)give(4);
  float*    consts = (float*)give(C_TOTAL * sizeof(float));
  float*    msg    = (float*)give((size_t)NV * MSTR * sizeof(float));
  float*    selfb  = (float*)give((size_t)NV * MSTR * sizeof(float));
  float*    aggc   = (float*)give((size_t)NC * MSTR * sizeof(float));
  int*      degv   = (int*)give((size_t)NV * sizeof(int));
  int*      degc   = (int*)give((size_t)NC * sizeof(int));
  // pooling block: [poolVs|poolCs|poolVt|poolCt|cntVs|cntCs|cntVt|cntCt]
  char*  poolblk = give((4 * MAXB * 17 + 4 * MAXB) * sizeof(float));
  float* poolVs = (float*)poolblk;
  float* poolCs = poolVs + MAXB * 17;
  float* poolVt = poolCs + MAXB * 17;
  float* poolCt = poolVt + MAXB * 17;
  int*   cntVs  = (int*)(poolCt + MAXB * 17);
  int*   cntCs  = cntVs + MAXB;
  int*   cntVt  = cntCs + MAXB;
  int*   cntCt  = cntVt + MAXB;

  // ---- constants (once; shared by both graphs) ----
  kprep<<<1, 256, 0, stream>>>(cons_ln_b, c1W, c1b, c2W, c2b, edge_ln_b,
                               cv[0][0], cv[0][1], cv[0][2], cv[0][3], cv[0][4], cv[0][5],
                               cv[1][0], cv[1][1], cv[1][2], cv[1][3], cv[1][4], cv[1][5],
                               cv[2][0], cv[2][1], cv[2][2], cv[2][3], cv[2][4], cv[2][5],
                               vl1W, consts);
  hipMemsetAsync(poolblk, 0, (4 * MAXB * 17 + 4 * MAXB) * sizeof(float), stream);

  const int numTiles  = (NV + 15) / 16;
  const int gB        = (numTiles + 7) / 8;
  const int gNV       = (NV + 255) / 256;
  const int gNC       = (NC + 255) / 256;
  const int gNE       = (NE + 255) / 256;

  for (int side = 0; side < 2; ++side) {
    const float* vf     = (const float*)d_in[side ? 9 : 3];
    const int*   ei     = (const int*)d_in[side ? 7 : 1];
    const int*   cbatch = (const int*)d_in[side ? 10 : 4];
    const int*   vbatch = (const int*)d_in[side ? 11 : 5];
    float* poolV = side ? poolVt : poolVs;
    float* poolC = side ? poolCt : poolCs;
    int*   cntV  = side ? cntVt  : cntVs;
    int*   cntC  = side ? cntCt  : cntCs;

    hipMemsetAsync(gmaxp, 0, 4, stream);
    hipMemsetAsync(degv, 0, (size_t)NV * sizeof(int), stream);
    hipMemsetAsync(degc, 0, (size_t)NC * sizeof(int), stream);
    hipMemsetAsync(aggc, 0, (size_t)NC * MSTR * sizeof(float), stream);

    kgmax<<<gNV, 256, 0, stream>>>(vf, gmaxp, NV);
    kvarmlp<<<gB, 256, 0, stream>>>(vf, var_ln_g, var_ln_b, (const float*)gmaxp,
                                    consts, vl1b, vl2W, vl2b, msg, selfb, NV, numTiles);
    kedges<<<gNE, 256, 0, stream>>>(ei, msg, aggc, degv, degc, NE);
    kcons<<<gNC, 256, 0, stream>>>(aggc, degc, cbatch, consts, poolC, cntC, NC);
    kvars<<<gNV, 256, 0, stream>>>(selfb, degv, vbatch, consts, poolV, cntV, NV);
  }

  kfinal<<<1, 128, 0, stream>>>(poolVs, poolCs, cntVs, cntCs,
                                poolVt, poolCt, cntVt, cntCt,
                                fln_g, fln_b, f1W, f1b, f2W, ngp, (float*)d_out);
}